// Sinkhorn_word_29386166239351
// MI455X (gfx1250) — compile-verified
//
#include <hip/hip_runtime.h>
#include <hip/hip_bf16.h>
#include <stdint.h>

// ---------------------------------------------------------------------------
// Sinkhorn loss, structure-exploiting form.
//   K = (1-q)I + q*ones, q = exp(-1/eps) = exp(-10)
//   x @ K.T  == (1-q)*x + q*sum(x)          (K symmetric, uniform row sums)
//   loss     == sum_rows q*(Su*Sv - <u,v>) / (S*B)
// One wave (32 lanes) owns one row of C=1024 floats (32 per lane, in VGPRs).
// Teacher row is staged global->LDS with the CDNA5 async path, overlapped
// with the student-side (u) Sinkhorn chain. Final cross-row reduction runs
// on the matrix pipe: sum == ones-matvec via V_WMMA_F32_16X16X4_F32.
// ---------------------------------------------------------------------------

#define C_DIM            1024
#define ELEMS_PER_LANE   32      // C / 32 lanes
#define F4_PER_LANE      8       // 16B chunks per lane
#define WAVES_PER_BLOCK  8
#define N_ITER           20

typedef __attribute__((ext_vector_type(2))) float v2f;
typedef __attribute__((ext_vector_type(8))) float v8f;

__device__ __constant__ const float Q_GIBBS = 4.53999297624848515e-05f; // exp(-10)

__device__ __forceinline__ float wred_add(float x) {
#pragma unroll
  for (int m = 16; m >= 1; m >>= 1) x += __shfl_xor(x, m, 32);
  return x;
}

__device__ __forceinline__ float wred_max(float x) {
#pragma unroll
  for (int m = 16; m >= 1; m >>= 1) x = fmaxf(x, __shfl_xor(x, m, 32));
  return x;
}

// softmax(logits / 2) over the row (32 elems/lane x 32 lanes), in place
__device__ __forceinline__ void softmax_half_temp(float* x) {
  float mx = x[0];
#pragma unroll
  for (int i = 1; i < ELEMS_PER_LANE; ++i) mx = fmaxf(mx, x[i]);
  mx = wred_max(mx);
  float se = 0.0f;
#pragma unroll
  for (int i = 0; i < ELEMS_PER_LANE; ++i) {
    x[i] = expf((x[i] - mx) * 0.5f);   // exp((l - max)/TEMP), TEMP = 2
    se += x[i];
  }
  se = wred_add(se);
  const float r = 1.0f / se;
#pragma unroll
  for (int i = 0; i < ELEMS_PER_LANE; ++i) x[i] *= r;
}

// 20 Sinkhorn iterations for one scaling vector w against marginal p.
//   w := s / ((1-q)*t + q*sum(t)),  t = p / ((1-q)*w + q*sum(w))
__device__ __forceinline__ void sinkhorn_chain(float* w, const float* p) {
  const float q     = Q_GIBBS;
  const float one_q = 1.0f - q;
  const float srow  = 1.0f + (float)(C_DIM - 1) * q;   // uniform K row sum
#pragma unroll
  for (int i = 0; i < ELEMS_PER_LANE; ++i) w[i] = 1.0f;
#pragma unroll 1
  for (int it = 0; it < N_ITER; ++it) {
    float Sw = 0.0f;
#pragma unroll
    for (int i = 0; i < ELEMS_PER_LANE; ++i) Sw += w[i];
    Sw = wred_add(Sw);
    const float qSw = q * Sw;
    float t[ELEMS_PER_LANE];
    float St = 0.0f;
#pragma unroll
    for (int i = 0; i < ELEMS_PER_LANE; ++i) {
      t[i] = p[i] / fmaf(one_q, w[i], qSw);
      St += t[i];
    }
    St = wred_add(St);
    const float qSt = q * St;
#pragma unroll
    for (int i = 0; i < ELEMS_PER_LANE; ++i)
      w[i] = srow / fmaf(one_q, t[i], qSt);
  }
}

__global__ __launch_bounds__(256) void sinkhorn_rows_kernel(
    const float* __restrict__ teacher, const float* __restrict__ student,
    float* __restrict__ row_out, int nrows) {
  const int lane = threadIdx.x & 31;
  const int wave = threadIdx.x >> 5;
  const int row  = blockIdx.x * WAVES_PER_BLOCK + wave;

  __shared__ float4 ldsA[WAVES_PER_BLOCK][C_DIM / 4];   // 32 KB

  if (row >= nrows) return;   // wave-uniform

  const float4* bg = (const float4*)(student + (size_t)row * C_DIM);
  const float4* ag = (const float4*)(teacher + (size_t)row * C_DIM);

  // ---- CDNA5 async copy: teacher row -> LDS, overlapped with u-chain ----
#pragma unroll
  for (int c = 0; c < F4_PER_LANE; ++c) {
    // low 32 bits of a generic pointer into LDS == LDS byte offset
    uint32_t loff = (uint32_t)(uintptr_t)(const void*)&ldsA[wave][c * 32 + lane];
    uint64_t ga   = (uint64_t)(uintptr_t)(const void*)&ag[c * 32 + lane];
    asm volatile("global_load_async_to_lds_b128 %0, %1, off"
                 :: "v"(loff), "v"(ga) : "memory");
  }

  float u[ELEMS_PER_LANE];
  {
    float bb[ELEMS_PER_LANE];
#pragma unroll
    for (int c = 0; c < F4_PER_LANE; ++c) {
      float4 t4 = bg[c * 32 + lane];                    // coalesced b128
      bb[4 * c + 0] = t4.x; bb[4 * c + 1] = t4.y;
      bb[4 * c + 2] = t4.z; bb[4 * c + 3] = t4.w;
    }
    softmax_half_temp(bb);          // b = softmax(student/2)
    sinkhorn_chain(u, bb);          // u-chain (teacher row still in flight)
  }

  // teacher row has had the whole u-chain to land in LDS
  asm volatile("s_wait_asynccnt 0x0" ::: "memory");

  float v[ELEMS_PER_LANE];
  {
    float aa[ELEMS_PER_LANE];
#pragma unroll
    for (int c = 0; c < F4_PER_LANE; ++c) {
      float4 t4 = ldsA[wave][c * 32 + lane];            // ds_load_b128
      aa[4 * c + 0] = t4.x; aa[4 * c + 1] = t4.y;
      aa[4 * c + 2] = t4.z; aa[4 * c + 3] = t4.w;
    }
    softmax_half_temp(aa);          // a = softmax(teacher/2)
    sinkhorn_chain(v, aa);          // v-chain
  }

  // per_row = q * (Su*Sv - <u,v>)   since M = K*cost = q*(ones - I)
  float Su = 0.0f, Sv = 0.0f, duv = 0.0f;
#pragma unroll
  for (int i = 0; i < ELEMS_PER_LANE; ++i) {
    Su += u[i]; Sv += v[i]; duv += u[i] * v[i];
  }
  Su = wred_add(Su); Sv = wred_add(Sv); duv = wred_add(duv);
  if (lane == 0) row_out[row] = Q_GIBBS * (Su * Sv - duv);
}

// ---------------------------------------------------------------------------
// Final reduction on the matrix pipe: sum(vals) == ones-matvec.
// D(16x16) = A(16x4) x ones(4x16) + C accumulates 64 values per WMMA with a
// hardware-fixed order (deterministic across graph replays). Single wave,
// EXEC all ones (WMMA requirement).
// C/D layout: VGPR r, lanes 0-15 -> D[M=r, N=lane]; lanes 16-31 -> D[M=r+8].
// Column 0 lives on lanes {0,16}: total = sum_r c[r] on both, then one
// shfl_xor(16) combine.
// ---------------------------------------------------------------------------
__global__ __launch_bounds__(32) void sinkhorn_reduce_wmma_kernel(
    const float* __restrict__ row_vals, float* __restrict__ out,
    int n, float inv_scale) {
  const int lane = threadIdx.x;           // single wave of 32, all active
  v8f c = {};                             // zero accumulator
  v2f ones;
  ones[0] = 1.0f; ones[1] = 1.0f;         // B = ones(4x16): 2 VGPRs of 1.0f
#pragma unroll 1
  for (int base = 0; base < n; base += 64) {
    const int i0 = base + lane;
    const int i1 = base + 32 + lane;
    v2f a;                                // A = next 64 values (16x4 tile)
    a[0] = (i0 < n) ? row_vals[i0] : 0.0f;
    a[1] = (i1 < n) ? row_vals[i1] : 0.0f;
    // 8 args: (neg_a, A, neg_b, B, c_mod, C, reuse_a, reuse_b)
    c = __builtin_amdgcn_wmma_f32_16x16x4_f32(
        false, a, false, ones, (short)0, c, false, false);
  }
  float s = c[0] + c[1] + c[2] + c[3] + c[4] + c[5] + c[6] + c[7];
  s += __shfl_xor(s, 16, 32);             // lane0: sum over all M of D[M,0]
  if (lane == 0) out[0] = s * inv_scale;
}

extern "C" void kernel_launch(void* const* d_in, const int* in_sizes, int n_in,
                              void* d_out, int out_size, void* d_ws, size_t ws_size,
                              hipStream_t stream) {
  const float* teacher = (const float*)d_in[0];   // [B,S,C] f32
  const float* student = (const float*)d_in[1];   // [B,S,C] f32
  const int total = in_sizes[0];
  const int nrows = total / C_DIM;                // B*S = 4096

  float* rows = (float*)d_ws;                     // nrows floats of scratch

  const int blocks = (nrows + WAVES_PER_BLOCK - 1) / WAVES_PER_BLOCK;
  sinkhorn_rows_kernel<<<blocks, 256, 0, stream>>>(teacher, student, rows, nrows);
  sinkhorn_reduce_wmma_kernel<<<1, 32, 0, stream>>>(rows, (float*)d_out, nrows,
                                                    1.0f / (float)nrows);
}